// GATNet_50740743635391
// MI455X (gfx1250) — compile-verified
//
#include <hip/hip_runtime.h>
#include <math.h>

typedef __attribute__((ext_vector_type(2))) float v2f;
typedef __attribute__((ext_vector_type(8))) float v8f;

#define NEG_SLOPE 0.2f

__device__ __forceinline__ float lrelu(float x) {
  return x > 0.0f ? x : NEG_SLOPE * x;
}

// ---------------------------------------------------------------------------
// Build extended B matrix [K][16]: cols 0..F-1 = W (K x F), col 8 = W @ a_src,
// col 9 = W @ a_dst, other cols zero. (s_src = h.a_src = x @ (W a_src))
// ---------------------------------------------------------------------------
__global__ void prep_wext_kernel(const float* __restrict__ W,
                                 const float* __restrict__ asrc,
                                 const float* __restrict__ adst,
                                 float* __restrict__ Wext, int K, int F) {
  int i = blockIdx.x * blockDim.x + threadIdx.x;
  if (i >= K * 16) return;
  int k = i >> 4;
  int n = i & 15;
  float v = 0.0f;
  if (n < F) {
    v = W[k * F + n];
  } else if (n == 8) {
    for (int j = 0; j < F; ++j) v += W[k * F + j] * asrc[j];
  } else if (n == 9) {
    for (int j = 0; j < F; ++j) v += W[k * F + j] * adst[j];
  }
  Wext[k * 16 + n] = v;
}

// ---------------------------------------------------------------------------
// Init per-node accumulators: m = -inf (segment max), z = 0, agg[F] = 0.
// ---------------------------------------------------------------------------
__global__ void init_node_kernel(float* __restrict__ m, float* __restrict__ z,
                                 float* __restrict__ agg, int N, int F) {
  int i = blockIdx.x * blockDim.x + threadIdx.x;
  if (i >= N) return;
  m[i] = -INFINITY;
  z[i] = 0.0f;
  for (int f = 0; f < F; ++f) agg[i * F + f] = 0.0f;
}

// ---------------------------------------------------------------------------
// WMMA GEMM: H[N,16] = A[N,K] @ Bext[K,16] using V_WMMA_F32_16X16X4_F32.
// One wave computes a 16x16 output tile (16 rows of A). K4 = K/4 steps.
// A fragment (16x4 f32): lanes 0-15 hold M=lane, VGPR0/1 = K=kb+0/kb+1;
// lanes 16-31 hold M=lane-16, VGPR0/1 = K=kb+2/kb+3.
// B fragment (4x16 f32): mirrored (N striped across lanes).
// C/D: VGPR r: lanes 0-15 -> (M=r, N=lane); lanes 16-31 -> (M=r+8, N=lane-16).
// ---------------------------------------------------------------------------
template <int K4>
__global__ void gemm_attn_wmma(const float* __restrict__ A, int lda,
                               const float* __restrict__ Bext,
                               float* __restrict__ H, int nRows) {
  int lane = threadIdx.x & 31;
  int wave = (int)((blockIdx.x * blockDim.x + threadIdx.x) >> 5);
  int row0 = wave * 16;
  if (row0 >= nRows) return;  // wave-uniform guard: EXEC stays all-1s

  int hf = lane >> 4;   // 0: lanes 0-15, 1: lanes 16-31
  int l  = lane & 15;

  v8f c = {};
#pragma unroll
  for (int k = 0; k < K4; ++k) {
    int kb = k * 4;
    const float* ap = A + (size_t)(row0 + l) * lda + kb + hf * 2;
    v2f a;
    a.x = ap[0];
    a.y = ap[1];
    const float* bp = Bext + (size_t)(kb + hf * 2) * 16 + l;
    v2f b;
    b.x = bp[0];
    b.y = bp[16];
    c = __builtin_amdgcn_wmma_f32_16x16x4_f32(false, a, false, b, (short)0, c,
                                              false, false);
  }

  float* out = H + (size_t)(row0 + hf * 8) * 16 + l;
#pragma unroll
  for (int r = 0; r < 8; ++r) out[(size_t)r * 16] = c[r];
}

// ---------------------------------------------------------------------------
// Edge pass 1: e = leaky_relu(s_src[src] + s_dst[dst]); m[dst] = max(m, e).
// s_src/s_dst live in H cols 8/9.
// ---------------------------------------------------------------------------
__global__ void edge_max_kernel(const int* __restrict__ src,
                                const int* __restrict__ dst,
                                const float* __restrict__ H,
                                float* __restrict__ m, int E) {
  int i = blockIdx.x * blockDim.x + threadIdx.x;
  if (i >= E) return;
  int s = src[i], d = dst[i];
  float e = lrelu(H[(size_t)s * 16 + 8] + H[(size_t)d * 16 + 9]);
  atomicMax(m + d, e);  // GLOBAL_ATOMIC_MAX_NUM_F32, resolves in L2
}

// ---------------------------------------------------------------------------
// Edge pass 2: p = exp(e - m[dst]); z[dst] += p. (e recomputed: 2 L2 floats
// beats spilling 6.4M edge scores.)
// ---------------------------------------------------------------------------
__global__ void edge_expsum_kernel(const int* __restrict__ src,
                                   const int* __restrict__ dst,
                                   const float* __restrict__ H,
                                   const float* __restrict__ m,
                                   float* __restrict__ z, int E) {
  int i = blockIdx.x * blockDim.x + threadIdx.x;
  if (i >= E) return;
  int s = src[i], d = dst[i];
  float e = lrelu(H[(size_t)s * 16 + 8] + H[(size_t)d * 16 + 9]);
  float p = __expf(e - m[d]);
  atomicAdd(z + d, p);
}

// ---------------------------------------------------------------------------
// Edge pass 3: alpha = p / (z[dst]+eps); agg[dst] += alpha * h[src].
// ---------------------------------------------------------------------------
template <int F>
__global__ void edge_agg_kernel(const int* __restrict__ src,
                                const int* __restrict__ dst,
                                const float* __restrict__ H,
                                const float* __restrict__ m,
                                const float* __restrict__ z,
                                float* __restrict__ agg, int E) {
  int i = blockIdx.x * blockDim.x + threadIdx.x;
  if (i >= E) return;
  int s = src[i], d = dst[i];
  float e = lrelu(H[(size_t)s * 16 + 8] + H[(size_t)d * 16 + 9]);
  float alpha = __expf(e - m[d]) / (z[d] + 1e-16f);
  const float* hs = H + (size_t)s * 16;
  float* o = agg + (size_t)d * F;
#pragma unroll
  for (int f = 0; f < F; ++f) atomicAdd(o + f, alpha * hs[f]);
}

// ---------------------------------------------------------------------------
// Layer-1 epilogue: h1r = relu(agg1 + b1), stored [N,8] for the next GEMM.
// ---------------------------------------------------------------------------
__global__ void relu_bias_kernel(const float* __restrict__ agg,
                                 const float* __restrict__ b,
                                 float* __restrict__ out, int N) {
  int i = blockIdx.x * blockDim.x + threadIdx.x;
  if (i >= N) return;
#pragma unroll
  for (int f = 0; f < 8; ++f) {
    float v = agg[(size_t)i * 8 + f] + b[f];
    out[(size_t)i * 8 + f] = v > 0.0f ? v : 0.0f;
  }
}

// ---------------------------------------------------------------------------
// Final: out = log_softmax(agg2 + b2) over 2 classes.
// ---------------------------------------------------------------------------
__global__ void logsoftmax_kernel(const float* __restrict__ agg,
                                  const float* __restrict__ b,
                                  float* __restrict__ out, int N) {
  int i = blockIdx.x * blockDim.x + threadIdx.x;
  if (i >= N) return;
  float o0 = agg[(size_t)i * 2 + 0] + b[0];
  float o1 = agg[(size_t)i * 2 + 1] + b[1];
  float mx = fmaxf(o0, o1);
  float lse = mx + __logf(__expf(o0 - mx) + __expf(o1 - mx));
  out[(size_t)i * 2 + 0] = o0 - lse;
  out[(size_t)i * 2 + 1] = o1 - lse;
}

// ---------------------------------------------------------------------------
// Host-side orchestration.
// ---------------------------------------------------------------------------
extern "C" void kernel_launch(void* const* d_in, const int* in_sizes, int n_in,
                              void* d_out, int out_size, void* d_ws,
                              size_t ws_size, hipStream_t stream) {
  const float* x      = (const float*)d_in[0];   // [N,36]
  const float* W1     = (const float*)d_in[1];   // [36,8]
  const float* a1_src = (const float*)d_in[2];   // [8]
  const float* a1_dst = (const float*)d_in[3];   // [8]
  const float* b1     = (const float*)d_in[4];   // [8]
  const float* W2     = (const float*)d_in[5];   // [8,2]
  const float* a2_src = (const float*)d_in[6];   // [2]
  const float* a2_dst = (const float*)d_in[7];   // [2]
  const float* b2     = (const float*)d_in[8];   // [2]
  const int*   eidx   = (const int*)d_in[9];     // [2,E]

  const int N = in_sizes[0] / 36;
  const int E = in_sizes[9] / 2;
  const int* src = eidx;
  const int* dst = eidx + E;

  // Workspace carve-up (floats)
  float* ws    = (float*)d_ws;
  float* h1    = ws;                 // [N,16] cols 0-7 h, 8 s_src, 9 s_dst
  float* m1    = h1 + (size_t)N * 16;
  float* z1    = m1 + N;
  float* agg1  = z1 + N;             // [N,8]
  float* h1r   = agg1 + (size_t)N * 8;   // [N,8]
  float* h2    = h1r + (size_t)N * 8;    // [N,16]
  float* m2    = h2 + (size_t)N * 16;
  float* z2    = m2 + N;
  float* agg2  = z2 + N;             // [N,2]
  float* wext1 = agg2 + (size_t)N * 2;   // [36,16]
  float* wext2 = wext1 + 36 * 16;        // [8,16]

  const int BLK = 256;
  dim3 nodeGrid((N + BLK - 1) / BLK);
  dim3 edgeGrid((E + BLK - 1) / BLK);
  // 16 rows per wave, 8 waves per 256-thread block -> 128 rows/block
  dim3 gemmGrid((N + 127) / 128);

  // ---- Layer 1 ----
  prep_wext_kernel<<<dim3((36 * 16 + BLK - 1) / BLK), BLK, 0, stream>>>(
      W1, a1_src, a1_dst, wext1, 36, 8);
  init_node_kernel<<<nodeGrid, BLK, 0, stream>>>(m1, z1, agg1, N, 8);
  gemm_attn_wmma<9><<<gemmGrid, BLK, 0, stream>>>(x, 36, wext1, h1, N);
  edge_max_kernel<<<edgeGrid, BLK, 0, stream>>>(src, dst, h1, m1, E);
  edge_expsum_kernel<<<edgeGrid, BLK, 0, stream>>>(src, dst, h1, m1, z1, E);
  edge_agg_kernel<8><<<edgeGrid, BLK, 0, stream>>>(src, dst, h1, m1, z1, agg1, E);
  relu_bias_kernel<<<nodeGrid, BLK, 0, stream>>>(agg1, b1, h1r, N);

  // ---- Layer 2 ----
  prep_wext_kernel<<<dim3(1), BLK, 0, stream>>>(W2, a2_src, a2_dst, wext2, 8, 2);
  init_node_kernel<<<nodeGrid, BLK, 0, stream>>>(m2, z2, agg2, N, 2);
  gemm_attn_wmma<2><<<gemmGrid, BLK, 0, stream>>>(h1r, 8, wext2, h2, N);
  edge_max_kernel<<<edgeGrid, BLK, 0, stream>>>(src, dst, h2, m2, E);
  edge_expsum_kernel<<<edgeGrid, BLK, 0, stream>>>(src, dst, h2, m2, z2, E);
  edge_agg_kernel<2><<<edgeGrid, BLK, 0, stream>>>(src, dst, h2, m2, z2, agg2, E);

  // ---- Output ----
  logsoftmax_kernel<<<nodeGrid, BLK, 0, stream>>>(agg2, b2, (float*)d_out, N);
}